// GroupAttention_82489141887736
// MI455X (gfx1250) — compile-verified
//
#include <hip/hip_runtime.h>
#include <hip/hip_bf16.h>
#include <math.h>

#define BB 4
#define SS 2048
#define DD 1024

typedef __attribute__((ext_vector_type(2))) float v2f;
typedef __attribute__((ext_vector_type(4))) float v4f;
typedef __attribute__((ext_vector_type(8))) float v8f;
typedef __attribute__((ext_vector_type(4))) int   v4i;

#if __has_builtin(__builtin_amdgcn_global_load_async_to_lds_b128) && \
    __has_builtin(__builtin_amdgcn_s_wait_asynccnt)
#define HAS_ASYNC 1
#else
#define HAS_ASYNC 0
#endif

typedef __attribute__((address_space(1))) v4i gas_v4i;
typedef __attribute__((address_space(3))) v4i las_v4i;

// 16-byte global -> LDS copy (async on CDNA5, sync fallback otherwise)
__device__ __forceinline__ void copy16_g2l(const float* g, float* l) {
#if HAS_ASYNC
  __builtin_amdgcn_global_load_async_to_lds_b128((gas_v4i*)g, (las_v4i*)l, 0, 0);
#else
  *(v4f*)l = *(const v4f*)g;
#endif
}

__device__ __forceinline__ void async_wait_3() {
#if HAS_ASYNC
  __builtin_amdgcn_s_wait_asynccnt(3);
#endif
}
__device__ __forceinline__ void async_wait_0() {
#if HAS_ASYNC
  __builtin_amdgcn_s_wait_asynccnt(0);
#endif
}

// ---------------------------------------------------------------------------
// 1) LayerNorm (Bessel-corrected variance, denom = sqrt(var)+eps)
// ---------------------------------------------------------------------------
__global__ __launch_bounds__(256) void ln_kernel(const float* __restrict__ ctx,
                                                 const float* __restrict__ gamma,
                                                 const float* __restrict__ beta,
                                                 float* __restrict__ xo) {
  const int row = blockIdx.x;            // 0 .. B*S-1
  const int tid = threadIdx.x;           // 256 threads, 4 floats each -> D=1024
  const float4* cp = (const float4*)(ctx + (size_t)row * DD);
  float4 v = cp[tid];
  float s1 = v.x + v.y + v.z + v.w;
  float s2 = v.x * v.x + v.y * v.y + v.z * v.z + v.w * v.w;
  for (int off = 16; off > 0; off >>= 1) {        // wave32 reduction
    s1 += __shfl_down(s1, off, 32);
    s2 += __shfl_down(s2, off, 32);
  }
  __shared__ float w1[8], w2[8], mb[2];
  const int wv = tid >> 5, ln = tid & 31;
  if (ln == 0) { w1[wv] = s1; w2[wv] = s2; }
  __syncthreads();
  if (tid == 0) {
    float t1 = 0.f, t2 = 0.f;
    for (int i = 0; i < 8; ++i) { t1 += w1[i]; t2 += w2[i]; }
    float mean = t1 / (float)DD;
    float var = fmaxf((t2 - (float)DD * mean * mean) / (float)(DD - 1), 0.f);
    mb[0] = mean;
    mb[1] = 1.f / (__builtin_sqrtf(var) + 1e-6f);
  }
  __syncthreads();
  const float mean = mb[0], rs = mb[1];
  float4 g4 = ((const float4*)gamma)[tid];
  float4 b4 = ((const float4*)beta)[tid];
  float4 o;
  o.x = g4.x * (v.x - mean) * rs + b4.x;
  o.y = g4.y * (v.y - mean) * rs + b4.y;
  o.z = g4.z * (v.z - mean) * rs + b4.z;
  o.w = g4.w * (v.w - mean) * rs + b4.w;
  ((float4*)(xo + (size_t)row * DD))[tid] = o;
}

// ---------------------------------------------------------------------------
// 2) FP32 WMMA GEMM with double-buffered async global->LDS staging.
//    C[MxN] = A[MxK] * B[KxN]   (TRANSB: B[k][n] = Bm[n*K+k])
//    block = 256 thr = 8 waves; block tile 128(M) x 64(N); wave tile 64x16.
//    Stage depth KT=16; A staged at stride 20 floats (bank-conflict-free:
//    20*lane mod 64 permutes banks since gcd(5,16)=1).
// ---------------------------------------------------------------------------
#define KT 16
#define AP 20   // padded A (and transposed-B) LDS row stride in floats

template <bool TRANSB>
__global__ __launch_bounds__(256) void wmma_gemm(const float* __restrict__ A,
                                                 const float* __restrict__ Bm,
                                                 float* __restrict__ C,
                                                 int M, int N, int K) {
  __shared__ float Ash[2][128 * AP];
  __shared__ float Bsh[2][TRANSB ? 64 * AP : KT * 64];

  const int tid  = threadIdx.x;
  const int wave = tid >> 5;
  const int lane = tid & 31;
  const int half = lane >> 4;
  const int lm   = lane & 15;
  const int bmB = blockIdx.y * 128;
  const int bnB = blockIdx.x * 64;
  const int mw = (wave >> 2) * 64;      // wave M offset within block tile
  const int nw = (wave & 3) * 16;       // wave N offset within block tile

  // per-thread stage-loader coordinates
  const int ar0 = (tid + 0)   >> 2, ac0 = ((tid + 0)   & 3) * 4;  // A piece 0
  const int ar1 = (tid + 256) >> 2, ac1 = ((tid + 256) & 3) * 4;  // A piece 1
  const int br  = TRANSB ? (tid >> 2) : (tid >> 4);
  const int bc  = (TRANSB ? (tid & 3) : (tid & 15)) * 4;

  v8f acc0 = {}, acc1 = {}, acc2 = {}, acc3 = {};

  const int NSTG = K / KT;
  // prologue: stage 0 into buffer 0
  {
    copy16_g2l(A + (size_t)(bmB + ar0) * K + ac0, &Ash[0][ar0 * AP + ac0]);
    copy16_g2l(A + (size_t)(bmB + ar1) * K + ac1, &Ash[0][ar1 * AP + ac1]);
    if (TRANSB)
      copy16_g2l(Bm + (size_t)(bnB + br) * K + bc, &Bsh[0][br * AP + bc]);
    else
      copy16_g2l(Bm + (size_t)br * N + bnB + bc, &Bsh[0][br * 64 + bc]);
  }

  for (int s = 0; s < NSTG; ++s) {
    const int cur = s & 1, nxt = cur ^ 1;
    const int ksn = (s + 1) * KT;
    if (s + 1 < NSTG) {           // kick off next stage while computing this one
      copy16_g2l(A + (size_t)(bmB + ar0) * K + ksn + ac0, &Ash[nxt][ar0 * AP + ac0]);
      copy16_g2l(A + (size_t)(bmB + ar1) * K + ksn + ac1, &Ash[nxt][ar1 * AP + ac1]);
      if (TRANSB)
        copy16_g2l(Bm + (size_t)(bnB + br) * K + ksn + bc, &Bsh[nxt][br * AP + bc]);
      else
        copy16_g2l(Bm + (size_t)(ksn + br) * N + bnB + bc, &Bsh[nxt][br * 64 + bc]);
      if (s + 2 < NSTG)           // far prefetch into L2 (global_prefetch_b8)
        __builtin_prefetch(A + (size_t)(bmB + ar0) * K + ksn + KT + ac0, 0, 2);
      async_wait_3();             // stage s resident, stage s+1 in flight
    } else {
      async_wait_0();
    }
    __syncthreads();

#pragma unroll
    for (int kk = 0; kk < KT; kk += 4) {
      const int k0 = kk + 2 * half;
      v2f a_0 = *(const v2f*)&Ash[cur][(mw +  0 + lm) * AP + k0];
      v2f a_1 = *(const v2f*)&Ash[cur][(mw + 16 + lm) * AP + k0];
      v2f a_2 = *(const v2f*)&Ash[cur][(mw + 32 + lm) * AP + k0];
      v2f a_3 = *(const v2f*)&Ash[cur][(mw + 48 + lm) * AP + k0];
      v2f bf;
      if (TRANSB) {
        bf = *(const v2f*)&Bsh[cur][(nw + lm) * AP + k0];
      } else {
        bf.x = Bsh[cur][(k0 + 0) * 64 + nw + lm];
        bf.y = Bsh[cur][(k0 + 1) * 64 + nw + lm];
      }
      acc0 = __builtin_amdgcn_wmma_f32_16x16x4_f32(false, a_0, false, bf, (short)0, acc0, false, false);
      acc1 = __builtin_amdgcn_wmma_f32_16x16x4_f32(false, a_1, false, bf, (short)0, acc1, false, false);
      acc2 = __builtin_amdgcn_wmma_f32_16x16x4_f32(false, a_2, false, bf, (short)0, acc2, false, false);
      acc3 = __builtin_amdgcn_wmma_f32_16x16x4_f32(false, a_3, false, bf, (short)0, acc3, false, false);
    }
    __syncthreads();              // all waves done reading before buffer reuse
  }

  const int col = bnB + nw + lm;
#pragma unroll
  for (int v = 0; v < 8; ++v) {
    const int mrow = bmB + mw + 8 * half + v;
    C[(size_t)(mrow +  0) * (size_t)N + col] = acc0[v];
    C[(size_t)(mrow + 16) * (size_t)N + col] = acc1[v];
    C[(size_t)(mrow + 32) * (size_t)N + col] = acc2[v];
    C[(size_t)(mrow + 48) * (size_t)N + col] = acc3[v];
  }
}

// ---------------------------------------------------------------------------
// 3) Aux vectors: v1 = Wk@bq, v2 = Wq@bk, bqbk = bq.bk
// ---------------------------------------------------------------------------
__global__ void aux_kernel(const float* __restrict__ Wq, const float* __restrict__ Wk,
                           const float* __restrict__ bq, const float* __restrict__ bk,
                           float* __restrict__ v1, float* __restrict__ v2,
                           float* __restrict__ bqbk) {
  int t = blockIdx.x * blockDim.x + threadIdx.x;
  if (t < DD) {
    float s = 0.f;
    for (int d = 0; d < DD; ++d) s += Wk[(size_t)t * DD + d] * bq[d];
    v1[t] = s;
  } else if (t < 2 * DD) {
    const int e = t - DD;
    float s = 0.f;
    for (int d = 0; d < DD; ++d) s += Wq[(size_t)e * DD + d] * bk[d];
    v2[e] = s;
  } else if (t == 2 * DD) {
    float s = 0.f;
    for (int d = 0; d < DD; ++d) s += bq[d] * bk[d];
    *bqbk = s;
  }
}

// ---------------------------------------------------------------------------
// 4) Band scores: s_sup[r]=score(i,i+1), s_sub[r]=score(i,i-1)
//    score(i,j) = (y_i.x_j + x_j.v1 + x_i.v2 + bqbk) / D
// ---------------------------------------------------------------------------
__global__ __launch_bounds__(256) void band_kernel(const float* __restrict__ X,
                                                   const float* __restrict__ Y,
                                                   const float* __restrict__ v1,
                                                   const float* __restrict__ v2,
                                                   const float* __restrict__ bqbk,
                                                   float* __restrict__ s_sup,
                                                   float* __restrict__ s_sub) {
  const int r = blockIdx.x;
  const int i = r & (SS - 1);
  const int tid = threadIdx.x;
  const int rn = (i < SS - 1) ? r + 1 : r;
  const int rp = (i > 0) ? r - 1 : r;
  float4 a = ((const float4*)(Y + (size_t)r  * DD))[tid];
  float4 b = ((const float4*)(X + (size_t)r  * DD))[tid];
  float4 n = ((const float4*)(X + (size_t)rn * DD))[tid];
  float4 p = ((const float4*)(X + (size_t)rp * DD))[tid];
  float4 u = ((const float4*)v1)[tid];
  float4 w = ((const float4*)v2)[tid];
  __shared__ float red[5][256];
  red[0][tid] = a.x * n.x + a.y * n.y + a.z * n.z + a.w * n.w;  // y_i . x_{i+1}
  red[1][tid] = a.x * p.x + a.y * p.y + a.z * p.z + a.w * p.w;  // y_i . x_{i-1}
  red[2][tid] = n.x * u.x + n.y * u.y + n.z * u.z + n.w * u.w;  // x_{i+1} . v1
  red[3][tid] = p.x * u.x + p.y * u.y + p.z * u.z + p.w * u.w;  // x_{i-1} . v1
  red[4][tid] = b.x * w.x + b.y * w.y + b.z * w.z + b.w * w.w;  // x_i . v2
  __syncthreads();
  for (int s = 128; s > 0; s >>= 1) {
    if (tid < s) {
#pragma unroll
      for (int c = 0; c < 5; ++c) red[c][tid] += red[c][tid + s];
    }
    __syncthreads();
  }
  if (tid == 0) {
    const float bb = *bqbk;
    const float inv = 1.f / (float)DD;
    s_sup[r] = (i < SS - 1) ? (red[0][0] + red[2][0] + red[4][0] + bb) * inv : 0.f;
    s_sub[r] = (i > 0)      ? (red[1][0] + red[3][0] + red[4][0] + bb) * inv : 0.f;
  }
}

// ---------------------------------------------------------------------------
// 5) 2-entry masked softmax per row + dead-row (uniform 1/S) detection.
// ---------------------------------------------------------------------------
__global__ void softband_kernel(const float* __restrict__ s_sup, const float* __restrict__ s_sub,
                                const int* __restrict__ eos,
                                float* __restrict__ psup, float* __restrict__ psub,
                                float* __restrict__ du) {
  const int r = blockIdx.x * blockDim.x + threadIdx.x;
  if (r >= BB * SS) return;
  const int b = r >> 11, i = r & (SS - 1);
  const int* erow = eos + (size_t)b * SS * SS + (size_t)i * SS;
  const bool aa = (i < SS - 1) && (erow[i + 1] != 0);
  const bool ac = (i > 0) && (erow[i - 1] != 0);
  const float U = 1.f / (float)SS;
  if (!aa && !ac) { psup[r] = U; psub[r] = U; du[r] = U; return; }
  const float va = aa ? s_sup[r] : -INFINITY;
  const float vc = ac ? s_sub[r] : -INFINITY;
  const float m = fmaxf(va, vc);
  const float za = aa ? __expf(va - m) : 0.f;
  const float zc = ac ? __expf(vc - m) : 0.f;
  const float iz = 1.f / (za + zc);
  psup[r] = za * iz;
  psub[r] = zc * iz;
  du[r] = 0.f;
}

// ---------------------------------------------------------------------------
// 6) Per-batch prefix scan of L[i] = log(na_final[i,i+1] + 1e-9)
// ---------------------------------------------------------------------------
__global__ __launch_bounds__(256) void scan_kernel(const float* __restrict__ prior,
                                                   const float* __restrict__ psup,
                                                   const float* __restrict__ psub,
                                                   float* __restrict__ Cpre) {
  const int b = blockIdx.x;
  const int tid = threadIdx.x;
  __shared__ float Lsh[SS];
  __shared__ float Csh[SS];
  const size_t rb = (size_t)b * SS;
  for (int w = 0; w < SS / 256; ++w) {
    const int i = tid + w * 256;
    float L = 0.f;
    if (i < SS - 1) {
      const float pv  = prior[(size_t)b * SS * SS + (size_t)i * SS + (i + 1)];
      const float sym = __builtin_sqrtf(psup[rb + i] * psub[rb + i + 1] + 1e-9f);
      const float naf = pv + (1.f - pv) * sym;
      L = __logf(naf + 1e-9f);
    }
    Lsh[i] = L;
  }
  __syncthreads();
  if (tid == 0) {
    float acc = 0.f;
    for (int i = 0; i < SS; ++i) { Csh[i] = acc; acc += Lsh[i]; }
  }
  __syncthreads();
  for (int w = 0; w < SS / 256; ++w) {
    const int i = tid + w * 256;
    Cpre[rb + i] = Csh[i];
  }
}

// ---------------------------------------------------------------------------
// 7) Bandwidth phase: fill na and g_attn. Streaming data (prior in, na/g out)
//    uses non-temporal hints so 200MB of single-use traffic skips L2 retention.
// ---------------------------------------------------------------------------
__global__ __launch_bounds__(256) void fill_kernel(const float* __restrict__ prior,
                                                   const float* __restrict__ psup,
                                                   const float* __restrict__ psub,
                                                   const float* __restrict__ du,
                                                   const float* __restrict__ Cpre,
                                                   float* __restrict__ g_out,
                                                   float* __restrict__ na_out) {
  const int blk = blockIdx.x;            // B*S*2 blocks
  const int r = blk >> 1;                // b*S + i
  const int i = r & (SS - 1);
  const int b = r >> 11;
  const int j0 = (blk & 1) * 1024 + threadIdx.x * 4;
  const size_t rowbase = (size_t)r * SS;
  const size_t brow = (size_t)b * SS;
  const float du_i = du[r];
  const float ps_i = psup[r];
  const float pb_i = psub[r];
  const float C_i = Cpre[r];
  const v4f pr4 = __builtin_nontemporal_load((const v4f*)(prior + rowbase + j0));
  const v4f cj4 = *(const v4f*)(Cpre + brow + j0);
  const v4f dj4 = *(const v4f*)(du + brow + j0);
  v4f no, go;
#pragma unroll
  for (int t = 0; t < 4; ++t) {
    const int j = j0 + t;
    float pi, pj;
    if (j == i + 1)      { pi = ps_i; pj = psub[brow + j]; }   // na0[i,i+1]*na0[i+1,i]
    else if (j == i - 1) { pi = pb_i; pj = psup[brow + j]; }   // na0[i,i-1]*na0[i-1,i]
    else                 { pi = du_i; pj = dj4[t]; }           // off-band / diagonal
    const float p = pr4[t];
    const float sym = __builtin_sqrtf(pi * pj + 1e-9f);
    const float naf = p + (1.f - p) * sym;
    float ga;
    if (j == i) {
      ga = naf;                                                // diag: na value
    } else {
      const float dlt = (j > i) ? (cj4[t] - C_i) : (C_i - cj4[t]);
      ga = __expf(dlt) + 1e-9f;                                // g + g^T + 1e-9
    }
    no[t] = naf;
    go[t] = ga;
  }
  __builtin_nontemporal_store(no, (v4f*)(na_out + rowbase + j0));
  __builtin_nontemporal_store(go, (v4f*)(g_out + rowbase + j0));
}

// ---------------------------------------------------------------------------
extern "C" void kernel_launch(void* const* d_in, const int* in_sizes, int n_in,
                              void* d_out, int out_size, void* d_ws, size_t ws_size,
                              hipStream_t stream) {
  (void)in_sizes; (void)n_in; (void)out_size; (void)ws_size;
  const float* ctx   = (const float*)d_in[0];
  const int*   eos   = (const int*)d_in[1];
  const float* prior = (const float*)d_in[2];
  const float* Wq    = (const float*)d_in[3];
  const float* bq    = (const float*)d_in[4];
  const float* Wk    = (const float*)d_in[5];
  const float* bk    = (const float*)d_in[6];
  const float* gamma = (const float*)d_in[7];
  const float* beta  = (const float*)d_in[8];

  float* ws = (float*)d_ws;
  size_t off = 0;
  float* X    = ws + off; off += (size_t)BB * SS * DD;   // 33.5 MB
  float* Y    = ws + off; off += (size_t)BB * SS * DD;   // 33.5 MB
  float* Mm   = ws + off; off += (size_t)DD * DD;        // 4 MB
  float* v1   = ws + off; off += DD;
  float* v2   = ws + off; off += DD;
  float* bqbk = ws + off; off += 4;                      // keep 16B alignment
  float* s_sup = ws + off; off += (size_t)BB * SS;
  float* s_sub = ws + off; off += (size_t)BB * SS;
  float* psup  = ws + off; off += (size_t)BB * SS;
  float* psub  = ws + off; off += (size_t)BB * SS;
  float* du    = ws + off; off += (size_t)BB * SS;
  float* Cpre  = ws + off; off += (size_t)BB * SS;

  float* g_out  = (float*)d_out;
  float* na_out = g_out + (size_t)BB * SS * SS;

  ln_kernel<<<BB * SS, 256, 0, stream>>>(ctx, gamma, beta, X);
  // M = Wq @ Wk^T  (B accessed transposed)
  wmma_gemm<true><<<dim3(DD / 64, DD / 128), 256, 0, stream>>>(Wq, Wk, Mm, DD, DD, DD);
  aux_kernel<<<9, 256, 0, stream>>>(Wq, Wk, bq, bk, v1, v2, bqbk);
  // Y = X @ M
  wmma_gemm<false><<<dim3(DD / 64, (BB * SS) / 128), 256, 0, stream>>>(X, Mm, Y, BB * SS, DD, DD);
  band_kernel<<<BB * SS, 256, 0, stream>>>(X, Y, v1, v2, bqbk, s_sup, s_sub);
  softband_kernel<<<(BB * SS) / 256, 256, 0, stream>>>(s_sup, s_sub, eos, psup, psub, du);
  scan_kernel<<<BB, 256, 0, stream>>>(prior, psup, psub, Cpre);
  fill_kernel<<<BB * SS * 2, 256, 0, stream>>>(prior, psup, psub, du, Cpre, g_out, na_out);
}